// PromptCOD_89773406421072
// MI455X (gfx1250) — compile-verified
//
#include <hip/hip_runtime.h>
#include <float.h>

typedef __attribute__((ext_vector_type(2))) float v2f;
typedef __attribute__((ext_vector_type(4))) float v4f;
typedef __attribute__((ext_vector_type(8))) float v8f;

#define NB       8192          // batch rows
#define EMB      512
#define POOL     100
#define POOL_PAD 112           // 7 tiles of 16
#define TOPK     4
#define CLEN     8

// ---------------------------------------------------------------------------
// Kernel 1: L2-normalize clip_key rows into a 112-row zero-padded buffer.
// One wave per row; shfl_xor wave32 reduction.
// ---------------------------------------------------------------------------
__global__ __launch_bounds__(32)
void nk_normalize(const float* __restrict__ key, float* __restrict__ nk) {
    const int row  = blockIdx.x;      // 0..111
    const int lane = threadIdx.x;     // 0..31
    if (row >= POOL) {                // zero-fill padding rows
        for (int e = lane; e < EMB; e += 32) nk[row * EMB + e] = 0.0f;
        return;
    }
    float ss = 0.0f;
    for (int e = lane; e < EMB; e += 32) {
        const float v = key[row * EMB + e];
        ss += v * v;
    }
    #pragma unroll
    for (int off = 16; off > 0; off >>= 1) ss += __shfl_xor(ss, off, 32);
    const float inv = 1.0f / fmaxf(sqrtf(ss), 1e-12f);
    for (int e = lane; e < EMB; e += 32) nk[row * EMB + e] = key[row * EMB + e] * inv;
}

// ---------------------------------------------------------------------------
// Kernel 2: x = relu(xq @ W^T + bias)   (8192x512) = (8192x512)(512x512)
// fp32 WMMA 16x16x4. Block = 256 threads (8 waves) computing a 64x64 tile.
// K staged through LDS in 64-wide chunks. Each wave owns two 16x16 accums.
//
// 16x4 f32 A-frag layout (ISA 7.12.2): lanes 0-15 -> M=lane, K={k,k+1};
// lanes 16-31 -> M=lane-16, K={k+2,k+3}. B (4x16) is the symmetric map
// with N in place of M.
// ---------------------------------------------------------------------------
__global__ __launch_bounds__(256)
void gemm1_relu(const float* __restrict__ xq, const float* __restrict__ W,
                const float* __restrict__ bias, float* __restrict__ xout) {
    __shared__ float As[64][68];   // 68-float row stride: 16B-aligned, bank-spread
    __shared__ float Bs[64][68];

    const int tid  = threadIdx.x;
    const int wave = tid >> 5;         // 0..7
    const int lane = tid & 31;
    const int wm   = (wave & 3) * 16;  // M sub-tile: 0,16,32,48
    const int wn   = (wave >> 2) * 16; // N sub-tile: 0,16 (+32 for 2nd accum)
    const int gm0  = blockIdx.y * 64;
    const int gn0  = blockIdx.x * 64;

    const int row  = lane & 15;
    const int koff = (lane < 16) ? 0 : 2;

    v8f acc0 = {};
    v8f acc1 = {};

    for (int kk = 0; kk < EMB; kk += 64) {
        // ---- stage 64x64 slabs of A (xq rows) and B (W rows) into LDS ----
        #pragma unroll
        for (int r = 0; r < 4; ++r) {
            const int i  = tid + r * 256;     // 0..1023
            const int lr = i >> 4;            // LDS row 0..63
            const int c4 = (i & 15) * 4;      // float column 0..60
            *(v4f*)&As[lr][c4] = *(const v4f*)&xq[(size_t)(gm0 + lr) * EMB + kk + c4];
            *(v4f*)&Bs[lr][c4] = *(const v4f*)&W [(size_t)(gn0 + lr) * EMB + kk + c4];
        }
        // speculative prefetch of the next K slab (global_prefetch_b8)
        if (kk + 64 < EMB) {
            const int lr = tid >> 2;
            __builtin_prefetch(&xq[(size_t)(gm0 + lr) * EMB + kk + 64], 0, 0);
            __builtin_prefetch(&W [(size_t)(gn0 + lr) * EMB + kk + 64], 0, 0);
        }
        __syncthreads();

        // ---- 16 WMMA K-steps of 4 over the staged chunk ----
        #pragma unroll
        for (int k2 = 0; k2 < 64; k2 += 4) {
            const v2f a  = *(const v2f*)&As[wm + row]     [k2 + koff];
            const v2f b0 = *(const v2f*)&Bs[wn + row]     [k2 + koff];
            const v2f b1 = *(const v2f*)&Bs[wn + 32 + row][k2 + koff];
            acc0 = __builtin_amdgcn_wmma_f32_16x16x4_f32(
                       false, a, false, b0, (short)0, acc0, false, false);
            acc1 = __builtin_amdgcn_wmma_f32_16x16x4_f32(
                       false, a, false, b1, (short)0, acc1, false, false);
        }
        __syncthreads();
    }

    // ---- epilogue: bias + ReLU, scatter per 16x16 C/D layout ----
    const int n0    = gn0 + wn + (lane & 15);
    const float bs0 = bias[n0];
    const float bs1 = bias[n0 + 32];
    const int mhi   = (lane < 16) ? 0 : 8;
    #pragma unroll
    for (int r = 0; r < 8; ++r) {
        const int gRow = gm0 + wm + r + mhi;
        xout[(size_t)gRow * EMB + n0     ] = fmaxf(acc0[r] + bs0, 0.0f);
        xout[(size_t)gRow * EMB + n0 + 32] = fmaxf(acc1[r] + bs1, 0.0f);
    }
}

// ---------------------------------------------------------------------------
// Kernel 3: scores = x @ nK^T   (8192 x 112). One wave per 16x16 tile.
// Row-scaling by 1/||x|| is rank-preserving, so it is omitted (top-k only
// consumes these scores).
// ---------------------------------------------------------------------------
__global__ __launch_bounds__(32)
void gemm2_scores(const float* __restrict__ x, const float* __restrict__ nk,
                  float* __restrict__ scores) {
    const int m0   = blockIdx.y * 16;
    const int n0   = blockIdx.x * 16;
    const int lane = threadIdx.x;
    const int row  = lane & 15;
    const int koff = (lane < 16) ? 0 : 2;

    v8f acc = {};
    #pragma unroll 4
    for (int k = 0; k < EMB; k += 4) {
        const v2f a = *(const v2f*)&x [(size_t)(m0 + row) * EMB + k + koff];
        const v2f b = *(const v2f*)&nk[(size_t)(n0 + row) * EMB + k + koff];
        acc = __builtin_amdgcn_wmma_f32_16x16x4_f32(
                  false, a, false, b, (short)0, acc, false, false);
    }
    const int mhi = (lane < 16) ? 0 : 8;
    #pragma unroll
    for (int r = 0; r < 8; ++r)
        scores[(size_t)(m0 + r + mhi) * POOL_PAD + n0 + (lane & 15)] = acc[r];
}

// ---------------------------------------------------------------------------
// Kernel 4: stable top-4 per row (ascending scan + strict '>' insertion
// reproduces jax.lax.top_k's lowest-index-wins tie-breaking).
// ---------------------------------------------------------------------------
__global__ __launch_bounds__(256)
void topk4(const float* __restrict__ scores, int* __restrict__ idx) {
    const int b = blockIdx.x * blockDim.x + threadIdx.x;
    if (b >= NB) return;
    float v0 = -FLT_MAX, v1 = -FLT_MAX, v2 = -FLT_MAX, v3 = -FLT_MAX;
    int   i0 = 0, i1 = 0, i2 = 0, i3 = 0;
    const float* srow = scores + (size_t)b * POOL_PAD;
    for (int k = 0; k < POOL; ++k) {
        const float s = srow[k];
        if      (s > v0) { v3=v2; i3=i2; v2=v1; i2=i1; v1=v0; i1=i0; v0=s; i0=k; }
        else if (s > v1) { v3=v2; i3=i2; v2=v1; i2=i1; v1=s;  i1=k; }
        else if (s > v2) { v3=v2; i3=i2; v2=s;  i2=k; }
        else if (s > v3) { v3=s;  i3=k; }
    }
    int* d = idx + (size_t)b * TOPK;
    d[0] = i0; d[1] = i1; d[2] = i2; d[3] = i3;
}

// ---------------------------------------------------------------------------
// Kernel 5: gather Ck/Cv. c_pool is 1.6 MB (L2-resident reads); outputs are
// 512 MB of pure streaming -> 128-bit non-temporal stores.
// One block per (b, t): copies 8 rows x 512 floats (rows 0-3 -> Ck, 4-7 -> Cv).
// ---------------------------------------------------------------------------
__global__ __launch_bounds__(256)
void gather_ckcv(const float* __restrict__ c_pool, const int* __restrict__ idx,
                 float* __restrict__ Ck, float* __restrict__ Cv) {
    const int bt = blockIdx.x;            // 0 .. NB*TOPK-1
    const int b  = bt >> 2;
    const int t  = bt & 3;
    const int id = idx[(size_t)b * TOPK + t];

    const v4f* src = (const v4f*)(c_pool + (size_t)id * CLEN * EMB);
    v4f* dk = (v4f*)(Ck + ((size_t)b * (TOPK * CLEN / 2) + t * (CLEN / 2)) * EMB);
    v4f* dv = (v4f*)(Cv + ((size_t)b * (TOPK * CLEN / 2) + t * (CLEN / 2)) * EMB);

    // half-tile = 4 rows * 512 floats = 512 v4f
    for (int i = threadIdx.x; i < 512; i += 256) {
        __builtin_nontemporal_store(src[i],       &dk[i]);
        __builtin_nontemporal_store(src[512 + i], &dv[i]);
    }
}

// ---------------------------------------------------------------------------
extern "C" void kernel_launch(void* const* d_in, const int* in_sizes, int n_in,
                              void* d_out, int out_size, void* d_ws, size_t ws_size,
                              hipStream_t stream) {
    const float* x_block  = (const float*)d_in[0];  // (8192, 512)
    const float* x_query  = (const float*)d_in[1];  // (8192, 1, 512) contiguous
    const float* clip_key = (const float*)d_in[2];  // (100, 512)
    const float* W        = (const float*)d_in[3];  // (512, 512)
    const float* bias     = (const float*)d_in[4];  // (512,)
    const float* c_pool   = (const float*)d_in[5];  // (100, 8, 512)
    (void)in_sizes; (void)n_in; (void)out_size; (void)ws_size;

    // workspace layout (floats)
    float* nk     = (float*)d_ws;                       // 112*512
    float* xw     = nk + (size_t)POOL_PAD * EMB;        // 8192*512
    float* scores = xw + (size_t)NB * EMB;              // 8192*112
    int*   idx    = (int*)(scores + (size_t)NB * POOL_PAD); // 8192*4

    // output layout: Ck | Cv | x_block
    float* out = (float*)d_out;
    float* Ck  = out;
    float* Cv  = Ck + (size_t)NB * (TOPK * CLEN / 2) * EMB;     // +67,108,864
    float* xbo = Cv + (size_t)NB * (TOPK * CLEN / 2) * EMB;     // +134,217,728

    nk_normalize<<<POOL_PAD, 32, 0, stream>>>(clip_key, nk);

    gemm1_relu<<<dim3(EMB / 64, NB / 64), 256, 0, stream>>>(x_query, W, bias, xw);

    gemm2_scores<<<dim3(POOL_PAD / 16, NB / 16), 32, 0, stream>>>(xw, nk, scores);

    topk4<<<NB / 256, 256, 0, stream>>>(scores, idx);

    gather_ckcv<<<NB * TOPK, 256, 0, stream>>>(c_pool, idx, Ck, Cv);

    hipMemcpyAsync(xbo, x_block, (size_t)NB * EMB * sizeof(float),
                   hipMemcpyDeviceToDevice, stream);
}